// QuantGraphConv_47682726920997
// MI455X (gfx1250) — compile-verified
//
#include <hip/hip_runtime.h>

typedef __attribute__((ext_vector_type(2))) float v2f;
typedef __attribute__((ext_vector_type(8))) float v8f;

static constexpr int kNodes   = 100000;
static constexpr int kEdges   = 1600000;
static constexpr int kInDim   = 29;
static constexpr int kFanIn   = 32;   // 29 features + 3 coord deltas
static constexpr int kOutDim  = 64;
static constexpr unsigned kNegInfBits = 0xFF800000u;

// ---------------------------------------------------------------------------
// Pass 0: fill output with -inf bit pattern so the atomic max trick works.
// ---------------------------------------------------------------------------
__global__ __launch_bounds__(256) void gc_init(unsigned* __restrict__ out) {
  int i = blockIdx.x * 256 + threadIdx.x;
  if (i < kNodes * kOutDim) out[i] = kNegInfBits;
}

// ---------------------------------------------------------------------------
// Pass 2: nodes with no incoming edges stayed at -inf -> 0.0f (ref semantics).
// ---------------------------------------------------------------------------
__global__ __launch_bounds__(256) void gc_finalize(float* __restrict__ out) {
  int i = blockIdx.x * 256 + threadIdx.x;
  if (i < kNodes * kOutDim) {
    if (__float_as_uint(out[i]) == kNegInfBits) out[i] = 0.0f;
  }
}

// ---------------------------------------------------------------------------
// Pass 1: per-wave 16-edge x 64-channel tile.
//   msg tile (16x32 f32)  x  weight^T (32x64 f32)  via V_WMMA_F32_16X16X4_F32
//   K=32 -> 8 WMMA per 16x16 tile, 4 N-tiles -> 32 WMMA per wave.
// 1.6M edges / (8 waves * 16 edges) = 12500 blocks exactly: no tail, EXEC
// is all ones at every WMMA as the ISA requires.
// ---------------------------------------------------------------------------
__global__ __launch_bounds__(256, 2) void gc_edge_wmma(
    const float* __restrict__ node,
    const float* __restrict__ feat,
    const long long* __restrict__ edges,   // (E,2) int64: [dst, src]
    const float* __restrict__ weight,      // (64,32) row-major
    const float* __restrict__ gamma,
    const float* __restrict__ beta,
    const float* __restrict__ rmean,
    const float* __restrict__ rvar,
    float* __restrict__ out)               // (N,64)
{
  const int lane  = threadIdx.x & 31;
  const int wave  = threadIdx.x >> 5;
  const int g     = lane >> 4;        // half-wave (selects K pair / M group)
  const int m     = lane & 15;        // edge row for A, column for B/C/D
  const int ebase = (blockIdx.x * 8 + wave) * 16;

  // --- gather the message row owned by this lane (both half-waves gather the
  // --- same 16 rows; duplicate addresses hit L0) -----------------------------
  const int e = ebase + m;
  const long long dn_m = edges[2LL * e + 0];
  const long long sn_m = edges[2LL * e + 1];
  const float* frow = feat + sn_m * kInDim;
  const float* ns   = node + sn_m * 3;
  const float* nd   = node + dn_m * 3;
  const float d0 = ns[0] - nd[0];
  const float d1 = ns[1] - nd[1];
  const float d2 = ns[2] - nd[2];

  auto msg_elem = [&](int k) -> float {
    if (k < kInDim) return frow[k];
    if (k == 29)    return d0;
    if (k == 30)    return d1;
    return d2;
  };

  // A fragments, ISA 16x4 f32 layout: lane (g,m) holds A[m][4kk+2g], A[m][4kk+2g+1]
  v2f a[8];
#pragma unroll
  for (int kk = 0; kk < 8; ++kk) {
    const int k0 = 4 * kk + 2 * g;
    a[kk].x = msg_elem(k0);
    a[kk].y = msg_elem(k0 + 1);
  }

#pragma unroll
  for (int nt = 0; nt < 4; ++nt) {
    const int o = 16 * nt + m;                 // output channel for this lane
    const float* wrow = weight + o * kFanIn;   // B[k][o] = weight[o][k]

    v8f c = {};
#pragma unroll
    for (int kk = 0; kk < 8; ++kk) {
      // B fragment, 4x16 f32 layout: lane (g,n) holds B[4kk+2g][n], B[4kk+2g+1][n]
      v2f b;
      b.x = wrow[4 * kk + 2 * g];
      b.y = wrow[4 * kk + 2 * g + 1];
      c = __builtin_amdgcn_wmma_f32_16x16x4_f32(
              /*neg_a=*/false, a[kk], /*neg_b=*/false, b,
              /*c_mod=*/(short)0, c, /*reuse_a=*/false, /*reuse_b=*/false);
    }

    // BN affine for this channel (same for all 8 rows this lane holds)
    const float inv_std = gamma[o] * rsqrtf(rvar[o] + 1e-5f);
    const float mu = rmean[o];
    const float bt = beta[o];

    // D layout: VGPR r -> edge row (r + 8*g), column o. Scatter with the
    // signed-max / unsigned-min float-max atomic trick (init is -inf bits).
#pragma unroll
    for (int r = 0; r < 8; ++r) {
      const float val = (c[r] - mu) * inv_std + bt;
      const int er = ebase + r + 8 * g;
      const long long dn = edges[2LL * er];     // broadcast across 16 lanes -> L0 hit
      float* addr = out + dn * kOutDim + o;
      if (val >= 0.0f) {
        atomicMax((int*)addr, __float_as_int(val));
      } else {
        atomicMin((unsigned int*)addr, __float_as_uint(val));
      }
    }
  }
}

// ---------------------------------------------------------------------------
extern "C" void kernel_launch(void* const* d_in, const int* in_sizes, int n_in,
                              void* d_out, int out_size, void* d_ws, size_t ws_size,
                              hipStream_t stream) {
  (void)in_sizes; (void)n_in; (void)out_size; (void)d_ws; (void)ws_size;

  const float*     node   = (const float*)d_in[0];
  const float*     feat   = (const float*)d_in[1];
  const long long* edges  = (const long long*)d_in[2];
  const float*     weight = (const float*)d_in[3];
  const float*     gamma  = (const float*)d_in[4];
  const float*     beta   = (const float*)d_in[5];
  const float*     rmean  = (const float*)d_in[6];
  const float*     rvar   = (const float*)d_in[7];
  float*           out    = (float*)d_out;

  const int total = kNodes * kOutDim;
  gc_init<<<(total + 255) / 256, 256, 0, stream>>>((unsigned*)out);
  gc_edge_wmma<<<kEdges / 128, 256, 0, stream>>>(
      node, feat, edges, weight, gamma, beta, rmean, rvar, out);
  gc_finalize<<<(total + 255) / 256, 256, 0, stream>>>(out);
}